// AdaAug_79817672229347
// MI455X (gfx1250) — compile-verified
//
#include <hip/hip_runtime.h>
#include <math.h>

// Problem constants
#define NB   64
#define CH   3
#define HH   256
#define WW   256
#define TAP  7
#define PADR 3
#define TH   16                      // output rows per block
#define AROWS (TH + TAP - 1)         // 22 halo rows
#define PSTR 24                      // floats of params per image

// Dynamic LDS layout (floats)
#define A_OFF   0                               // colored tiles: 3 * AROWS * WW
#define B_OFF   (3 * AROWS * WW)                // h-conv buffer:   AROWS * WW
#define NZ_OFF  (B_OFF + AROWS * WW)            // noise tile:    3 * TH * WW
#define SMEM_FLOATS (NZ_OFF + 3 * TH * WW)      // 34816 floats
#define SMEM_BYTES  (SMEM_FLOATS * 4)           // 139264 B (2 blocks / WGP @ 320KB)

__device__ __forceinline__ void mat4mul(float* __restrict__ o,
                                        const float* __restrict__ a,
                                        const float* __restrict__ b) {
  #pragma unroll
  for (int i = 0; i < 4; ++i)
    #pragma unroll
    for (int j = 0; j < 4; ++j) {
      float acc = 0.0f;
      #pragma unroll
      for (int k = 0; k < 4; ++k) acc = fmaf(a[i*4+k], b[k*4+j], acc);
      o[i*4+j] = acc;
    }
}

// ---------------- Kernel 1: per-image parameters -> d_ws ----------------
__global__ void adaaug_params_kernel(const float* __restrict__ gates,
                                     const float* __restrict__ gauss,
                                     const float* __restrict__ unif,
                                     const float* __restrict__ fbank,
                                     float* __restrict__ ws) {
  int n = blockIdx.x * blockDim.x + threadIdx.x;
  if (n >= NB) return;
  const float P   = 1.0f;
  const float A3  = 0.57735026918962576f;   // 1/sqrt(3)
  const float PIc = 3.14159265358979323846f;

  float gt[11], gs[8], un[4];
  #pragma unroll
  for (int i = 0; i < 11; ++i) gt[i] = gates[n*11+i];
  #pragma unroll
  for (int i = 0; i < 8;  ++i) gs[i] = gauss[n*8+i];
  #pragma unroll
  for (int i = 0; i < 4;  ++i) un[i] = unif[n*4+i];

  // brightness + contrast:  Cm0 = S @ T
  float b  = (gt[0] < P) ? gs[0]*0.2f : 0.0f;
  float cc = (gt[1] < P) ? exp2f(gs[1]*0.5f) : 1.0f;
  float M0[16] = { cc,0,0,cc*b,  0,cc,0,cc*b,  0,0,cc,cc*b,  0,0,0,1 };

  // luma flip:  L = I4 - 2*vv*ilf   (vv top-left 3x3 = 1/3)
  float ilf = (gt[2] < P) ? floorf(un[0]*2.0f) : 0.0f;
  float L[16];
  #pragma unroll
  for (int i = 0; i < 16; ++i) L[i] = 0.0f;
  L[0] = L[5] = L[10] = L[15] = 1.0f;
  float d = -2.0f*(1.0f/3.0f)*ilf;
  #pragma unroll
  for (int i = 0; i < 3; ++i)
    #pragma unroll
    for (int j = 0; j < 3; ++j) L[i*4+j] += d;

  // hue rotation R4
  float theta = (gt[3] < P) ? (un[1]*2.0f - 1.0f)*PIc : 0.0f;
  float co = cosf(theta), si = sinf(theta);
  const float K[9] = {0,-A3,A3,  A3,0,-A3,  -A3,A3,0};
  float R[16];
  #pragma unroll
  for (int i = 0; i < 16; ++i) R[i] = 0.0f;
  R[15] = 1.0f;
  #pragma unroll
  for (int i = 0; i < 3; ++i)
    #pragma unroll
    for (int j = 0; j < 3; ++j)
      R[i*4+j] = (1.0f-co)*(1.0f/3.0f) + co*((i==j)?1.0f:0.0f) + si*K[i*3+j];

  // saturation:  Sm = vv + (I4 - vv)*s
  float s = (gt[4] < P) ? exp2f(gs[2]) : 1.0f;
  float Sm[16];
  #pragma unroll
  for (int i = 0; i < 4; ++i)
    #pragma unroll
    for (int j = 0; j < 4; ++j) {
      float vv = (i < 3 && j < 3) ? (1.0f/3.0f) : 0.0f;
      float I4 = (i == j) ? 1.0f : 0.0f;
      Sm[i*4+j] = vv + (I4 - vv)*s;
    }

  float T1[16], T2[16], T3[16];
  mat4mul(T1, L,  M0);
  mat4mul(T2, R,  T1);
  mat4mul(T3, Sm, T2);

  float* w = ws + n*PSTR;
  #pragma unroll
  for (int i = 0; i < 3; ++i)
    #pragma unroll
    for (int j = 0; j < 4; ++j) w[i*4+j] = T3[i*4+j];

  // 7-tap separable kernel
  const float ep[4] = {10.0f/13.0f, 1.0f/13.0f, 1.0f/13.0f, 1.0f/13.0f};
  float g[4] = {1,1,1,1};
  #pragma unroll
  for (int i = 0; i < 4; ++i) {
    float ti = (gt[5+i] < P) ? exp2f(gs[3+i]) : 1.0f;
    float t[4] = {1,1,1,1};
    t[i] = ti;
    float ssum = 0.0f;
    #pragma unroll
    for (int j = 0; j < 4; ++j) ssum += ep[j]*t[j]*t[j];
    float inv = 1.0f / sqrtf(ssum);
    #pragma unroll
    for (int j = 0; j < 4; ++j) g[j] *= t[j]*inv;
  }
  #pragma unroll
  for (int j = 0; j < TAP; ++j) {
    float kj = 0.0f;
    #pragma unroll
    for (int bnd = 0; bnd < 4; ++bnd) kj = fmaf(g[bnd], fbank[bnd*TAP+j], kj);
    w[12+j] = kj;
  }
  w[19] = (gt[9]  < P) ? fabsf(gs[7])*0.1f : 0.0f;   // sigma
  w[20] = ((gt[10] < P) ? 0.5f : 0.0f) * 0.5f;       // half cutout size
  w[21] = un[2];                                     // cx
  w[22] = un[3];                                     // cy
  w[23] = 0.0f;
}

__device__ __forceinline__ int reflect256(int p) {
  p = (p < 0) ? -p : p;
  p = (p > 255) ? (510 - p) : p;
  return p;
}

// ---------------- Kernel 2: fused color + separable conv + noise + cutout --------
__launch_bounds__(256)
__global__ void adaaug_main_kernel(const float* __restrict__ images,
                                   const float* __restrict__ noise,
                                   const float* __restrict__ params,
                                   float* __restrict__ out) {
  extern __shared__ float smem[];
  float* A  = smem + A_OFF;    // colored halo tiles (3 ch)
  float* B  = smem + B_OFF;    // h-conv scratch (1 ch)
  float* NZ = smem + NZ_OFF;   // noise tile (3 ch), filled asynchronously

  const int tid = threadIdx.x;
  const int r0  = blockIdx.x * TH;
  const int n   = blockIdx.y;

  // ---- Phase 0: issue async global->LDS copies of the noise tile (ASYNCcnt) ----
  // 3*TH*64 = 3072 float4 quads, 12 per thread, b128 per lane.
  #pragma unroll
  for (int i = 0; i < 12; ++i) {
    int q   = tid + i * 256;
    int c   = q >> 10;           // / (TH*64)
    int rem = q & 1023;
    int y   = rem >> 6;
    int qq  = rem & 63;
    int goff = ((((n*CH + c)*HH) + (r0 + y))*WW + qq*4) * 4;  // byte offset
    unsigned ldst = (unsigned)(unsigned long long)(uintptr_t)(const void*)
                    &NZ[(c*TH + y)*WW + qq*4];
    asm volatile("global_load_async_to_lds_b128 %0, %1, %2"
                 :
                 : "v"(ldst), "v"(goff), "s"(noise)
                 : "memory");
  }

  // ---- per-image parameters (uniform loads) ----
  const float* w = params + n*PSTR;
  const float M00=w[0],  M01=w[1],  M02=w[2],  M03=w[3];
  const float M10=w[4],  M11=w[5],  M12=w[6],  M13=w[7];
  const float M20=w[8],  M21=w[9],  M22=w[10], M23=w[11];
  float k7[TAP];
  #pragma unroll
  for (int j = 0; j < TAP; ++j) k7[j] = w[12+j];
  const float sigma=w[19], hsz=w[20], cx=w[21], cy=w[22];

  // ---- Phase 1: load raw image halo (3ch, float4) + color transform -> A ----
  // AROWS*64 = 1408 quads; iterations 0..4 are always in range, 5th is guarded.
  #pragma unroll
  for (int it = 0; it < 6; ++it) {
    int q = tid + it * 256;
    if (it < 5 || q < AROWS * 64) {
      int y  = q >> 6;
      int qq = q & 63;
      int gy = reflect256(r0 - PADR + y);
      const float4 p0 = *(const float4*)&images[(((n*CH+0)*HH)+gy)*WW + qq*4];
      const float4 p1 = *(const float4*)&images[(((n*CH+1)*HH)+gy)*WW + qq*4];
      const float4 p2 = *(const float4*)&images[(((n*CH+2)*HH)+gy)*WW + qq*4];
      float4 a0, a1, a2;
      a0.x = fmaf(M00,p0.x, fmaf(M01,p1.x, fmaf(M02,p2.x, M03)));
      a0.y = fmaf(M00,p0.y, fmaf(M01,p1.y, fmaf(M02,p2.y, M03)));
      a0.z = fmaf(M00,p0.z, fmaf(M01,p1.z, fmaf(M02,p2.z, M03)));
      a0.w = fmaf(M00,p0.w, fmaf(M01,p1.w, fmaf(M02,p2.w, M03)));
      a1.x = fmaf(M10,p0.x, fmaf(M11,p1.x, fmaf(M12,p2.x, M13)));
      a1.y = fmaf(M10,p0.y, fmaf(M11,p1.y, fmaf(M12,p2.y, M13)));
      a1.z = fmaf(M10,p0.z, fmaf(M11,p1.z, fmaf(M12,p2.z, M13)));
      a1.w = fmaf(M10,p0.w, fmaf(M11,p1.w, fmaf(M12,p2.w, M13)));
      a2.x = fmaf(M20,p0.x, fmaf(M21,p1.x, fmaf(M22,p2.x, M23)));
      a2.y = fmaf(M20,p0.y, fmaf(M21,p1.y, fmaf(M22,p2.y, M23)));
      a2.z = fmaf(M20,p0.z, fmaf(M21,p1.z, fmaf(M22,p2.z, M23)));
      a2.w = fmaf(M20,p0.w, fmaf(M21,p1.w, fmaf(M22,p2.w, M23)));
      *(float4*)&A[(0*AROWS + y)*WW + qq*4] = a0;
      *(float4*)&A[(1*AROWS + y)*WW + qq*4] = a1;
      *(float4*)&A[(2*AROWS + y)*WW + qq*4] = a2;
    }
  }

  // Drain this wave's async copies, then workgroup barrier makes A and NZ
  // visible to everyone.
#if __has_builtin(__builtin_amdgcn_s_wait_asynccnt)
  __builtin_amdgcn_s_wait_asynccnt(0);
#else
  asm volatile("s_wait_asynccnt 0" ::: "memory");
#endif
  __syncthreads();

  const int qx   = tid & 63;
  const int ysub = tid >> 6;

  // Hoisted: reflected horizontal tap columns (depend only on tid).
  int xoff[TAP];
  #pragma unroll
  for (int j = 0; j < TAP; ++j) xoff[j] = reflect256(tid + j - PADR);

  // Hoisted: per-column cutout keep-multipliers (depend only on qx).
  float mx[4];
  #pragma unroll
  for (int u = 0; u < 4; ++u) {
    float cox = ((float)(qx*4 + u) + 0.5f) * (1.0f/WW);
    mx[u] = (fabsf(cox - cx) >= hsz) ? 1.0f : 0.0f;
  }

  for (int c = 0; c < CH; ++c) {
    const float* Ac = A + c * AROWS * WW;

    // ---- Phase 2: horizontal 7-tap conv with reflect -> B (one thread per col)
    for (int y = 0; y < AROWS; ++y) {
      const float* row = Ac + y*WW;
      float acc = k7[0] * row[xoff[0]];
      #pragma unroll
      for (int j = 1; j < TAP; ++j) acc = fmaf(k7[j], row[xoff[j]], acc);
      B[y*WW + tid] = acc;
    }
    __syncthreads();

    // ---- Phase 3: vertical 7-tap conv + noise + cutout -> global (float4) ----
    #pragma unroll
    for (int yy = 0; yy < TH; yy += 4) {
      int y = yy + ysub;
      float4 acc = make_float4(0.f, 0.f, 0.f, 0.f);
      #pragma unroll
      for (int t = 0; t < TAP; ++t) {
        float4 bv = *(const float4*)&B[(y + t)*WW + qx*4];
        acc.x = fmaf(k7[t], bv.x, acc.x);
        acc.y = fmaf(k7[t], bv.y, acc.y);
        acc.z = fmaf(k7[t], bv.z, acc.z);
        acc.w = fmaf(k7[t], bv.w, acc.w);
      }
      float4 nz = *(const float4*)&NZ[(c*TH + y)*WW + qx*4];
      acc.x = fmaf(sigma, nz.x, acc.x);
      acc.y = fmaf(sigma, nz.y, acc.y);
      acc.z = fmaf(sigma, nz.z, acc.z);
      acc.w = fmaf(sigma, nz.w, acc.w);

      int gy = r0 + y;
      float coy = ((float)gy + 0.5f) * (1.0f/HH);
      // row-keep: if row is outside the cutout band, keep everything (mult 1)
      float myk = (fabsf(coy - cy) >= hsz) ? 1.0f : 0.0f;
      float4 res;
      res.x = acc.x * fmaxf(myk, mx[0]);
      res.y = acc.y * fmaxf(myk, mx[1]);
      res.z = acc.z * fmaxf(myk, mx[2]);
      res.w = acc.w * fmaxf(myk, mx[3]);

      *(float4*)&out[(((n*CH + c)*HH) + gy)*WW + qx*4] = res;
    }
    __syncthreads();   // B is reused by the next channel
  }
}

// ---------------- Host entry ----------------
extern "C" void kernel_launch(void* const* d_in, const int* in_sizes, int n_in,
                              void* d_out, int out_size, void* d_ws, size_t ws_size,
                              hipStream_t stream) {
  const float* images = (const float*)d_in[0];
  const float* gates  = (const float*)d_in[1];
  const float* gauss  = (const float*)d_in[2];
  const float* unif   = (const float*)d_in[3];
  const float* noise  = (const float*)d_in[4];
  const float* fbank  = (const float*)d_in[5];
  float* out = (float*)d_out;
  float* ws  = (float*)d_ws;

  // Allow 136 KB of dynamic LDS (WGP has 320 KB); idempotent, capture-safe.
  (void)hipFuncSetAttribute(reinterpret_cast<const void*>(&adaaug_main_kernel),
                            hipFuncAttributeMaxDynamicSharedMemorySize,
                            SMEM_BYTES);

  adaaug_params_kernel<<<1, 64, 0, stream>>>(gates, gauss, unif, fbank, ws);
  adaaug_main_kernel<<<dim3(HH/TH, NB), 256, SMEM_BYTES, stream>>>(images, noise, ws, out);
}